// AttentionLayer_1666447311232
// MI455X (gfx1250) — compile-verified
//
#include <hip/hip_runtime.h>

#ifndef __has_builtin
#define __has_builtin(x) 0
#endif

#define EPS 1e-6f

// ---------------- CDNA5 feature detection (device pass only) ----------------
#if defined(__gfx1250__)
#define IS_GFX1250 1
#else
#define IS_GFX1250 0
#endif

#if IS_GFX1250 && __has_builtin(__builtin_amdgcn_global_load_async_to_lds_b128)
#define ASYNC_VIA_BUILTIN 1
#else
#define ASYNC_VIA_BUILTIN 0
#endif

#if IS_GFX1250 && __has_builtin(__builtin_amdgcn_wmma_f32_16x16x4_f32)
#define USE_WMMA 1
#else
#define USE_WMMA 0
#endif

typedef __attribute__((ext_vector_type(2))) float v2f;
typedef __attribute__((ext_vector_type(8))) float v8f;
// 16-byte payload type expected by the async-LDS builtin (per hipcc diagnostic)
typedef __attribute__((__vector_size__(4 * sizeof(int)))) int i128_t;

// 16-byte async copy: global -> LDS (tracked by ASYNCcnt)
__device__ __forceinline__ void async_copy_b128(const float* g, float* s) {
#if IS_GFX1250
#if ASYNC_VIA_BUILTIN
  __builtin_amdgcn_global_load_async_to_lds_b128(
      (__attribute__((address_space(1))) i128_t*)(g),
      (__attribute__((address_space(3))) i128_t*)(s), 0, 0);
#else
  // lds dest address = low 32 bits of the generic shared pointer
  asm volatile("global_load_async_to_lds_b128 %0, %1, off"
               :: "v"((unsigned)(unsigned long long)s),
                  "v"((unsigned long long)g)
               : "memory");
#endif
#else
  *(float4*)s = *(const float4*)g;  // host / non-gfx1250 parse path
#endif
}

#if IS_GFX1250
#if __has_builtin(__builtin_amdgcn_s_wait_asynccnt)
#define WAIT_ASYNC(n) __builtin_amdgcn_s_wait_asynccnt(n)
#else
#define WAIT_ASYNC(n) asm volatile("s_wait_asynccnt %0" :: "n"(n) : "memory")
#endif
#else
#define WAIT_ASYNC(n) do { } while (0)
#endif

// B=8, L=2048, K=32, G=4, H=16, F=8 ; tokens = B*L = 16384
// One wave32 per token; 2 waves (64 threads) per block.
__global__ __launch_bounds__(64) void attn_fused(
    const float* __restrict__ g_beta,   // [T,64]   (G*H)
    const float* __restrict__ g_self,   // [T,64]
    const float* __restrict__ g_ac,     // [T,32,64]
    const float* __restrict__ g_nodes,  // [T,32,128] (F*H)
    const float* __restrict__ g_gw,     // [T,32,4]
    float* __restrict__ g_out,          // [T,128]
    float* __restrict__ g_acf)          // [T,32,16]
{
  __shared__ float s_ac[2][2048];   // 16 KB (reused as WMMA spill scratch later)
  __shared__ float s_nd[2][4096];   // 32 KB
  __shared__ float s_gw[2][128];    //  1 KB
  __shared__ float s_acf[2][512];   //  4 KB  (unnormalized acf[k,h])

  const int w  = threadIdx.x >> 5;
  const int l  = threadIdx.x & 31;
  const int t  = blockIdx.x * 2 + w;
  const int h  = l & 15;
  const int g0 = l >> 4;            // lane covers g0 and g0+2 for column h

  const float* ac_t = g_ac    + (size_t)t * 2048;
  const float* nd_t = g_nodes + (size_t)t * 4096;
  const float* gw_t = g_gw    + (size_t)t * 128;

  // ---- issue all tile transfers asynchronously (49 x B128 per wave) ----
#pragma unroll
  for (int i = 0; i < 16; ++i)                       // ac tile: 8 KB
    async_copy_b128(ac_t + (i * 32 + l) * 4, &s_ac[w][(i * 32 + l) * 4]);
  async_copy_b128(gw_t + l * 4, &s_gw[w][l * 4]);    // gw tile: 512 B
#pragma unroll
  for (int i = 0; i < 32; ++i)                       // nodes tile: 16 KB
    async_copy_b128(nd_t + (i * 32 + l) * 4, &s_nd[w][(i * 32 + l) * 4]);

  // small per-(g,h) params straight from global (coalesced)
  const float b0  = g_beta[(size_t)t * 64 + l]      + EPS;
  const float b1  = g_beta[(size_t)t * 64 + 32 + l] + EPS;
  const float sa0 = g_self[(size_t)t * 64 + l];
  const float sa1 = g_self[(size_t)t * 64 + 32 + l];

  // wait for ac+gw (first 17 retire in order); nodes tile keeps streaming
  WAIT_ASYNC(32);

  // ---- pass 1: max over (K,G) per h ----
  float m = -3.402823466e38f;
#pragma unroll
  for (int k = 0; k < 32; ++k) {
    float a0 = s_ac[w][k * 64 + l];
    float a1 = s_ac[w][k * 64 + 32 + l];
    if (k == 0) { a0 += sa0; a1 += sa1; }
    m = fmaxf(m, fmaxf(a0 * b0, a1 * b1));
  }
  m = fmaxf(m, __shfl_xor(m, 16, 32));   // fold g-pairs: full (K,G) max per h

  // ---- pass 2: s_acf[k,h] = sum_g gw * exp(..); S = abs-sum over K ----
  float S = 0.f;
#pragma unroll
  for (int k = 0; k < 32; ++k) {
    float a0 = s_ac[w][k * 64 + l];
    float a1 = s_ac[w][k * 64 + 32 + l];
    if (k == 0) { a0 += sa0; a1 += sa1; }
    const float w0 = s_gw[w][k * 4 + g0];
    const float w1 = s_gw[w][k * 4 + g0 + 2];
    float v = w0 * __expf(a0 * b0 - m) + w1 * __expf(a1 * b1 - m);
    v += __shfl_xor(v, 16, 32);          // full sum over 4 g, replicated on lane pair
    s_acf[w][k * 16 + h] = v;            // pair lanes write the same value
    S += fabsf(v);
  }
  const float rn = 1.0f / (S + EPS);     // per-h normalizer (folded into consumers)

  // ---- store normalized acf (16 fully-coalesced 128B stores) ----
  // flat LDS layout [k*16+h] matches the global layout; lane's rn matches its h
  float* acf_t = g_acf + (size_t)t * 512;
#pragma unroll
  for (int j = 0; j < 16; ++j) {
    const float v = s_acf[w][j * 32 + l] * rn;
    __builtin_nontemporal_store(v, acf_t + j * 32 + l);
  }

  // nodes tile must be resident now
  WAIT_ASYNC(0);

  float* out_t = g_out + (size_t)t * 128;

#if USE_WMMA
  // out[f,h] = sum_k nodes[k,f,h]*acf[k,h] = diag( N_f^T(16hx32k) x W(32kx16h) )
  // 8 chained V_WMMA_F32_16X16X4_F32 per f; FLOP overhead is free (BW-bound).
  v8f D[8] = {};
  const int kb    = (l >> 4) * 2;        // A: lanes 0-15 hold K=0,1; 16-31 hold K=2,3
  const int bbase = (l < 16) ? h : (h + 32);  // B row k or k+2, column h
#pragma unroll
  for (int c = 0; c < 8; ++c) {
    v2f bm;                              // B[kk,n] = rn(n)*acf[4c+kk, n]
    bm[0] = s_acf[w][(4 * c + 0) * 16 + bbase] * rn;
    bm[1] = s_acf[w][(4 * c + 1) * 16 + bbase] * rn;
    const int k0 = 4 * c + kb;
#pragma unroll
    for (int f = 0; f < 8; ++f) {
      v2f am;                            // A[m=h, kk] = nodes[4c+kk, f, h]
      am[0] = s_nd[w][(k0 + 0) * 128 + f * 16 + h];
      am[1] = s_nd[w][(k0 + 1) * 128 + f * 16 + h];
      D[f] = __builtin_amdgcn_wmma_f32_16x16x4_f32(
          false, am, false, bm, (short)0, D[f], false, false);
    }
  }

  // ---- diagonal extraction via LDS (s_ac[w] is dead: reuse 2048 floats) ----
  // lane l holds D[f][r] = element (M = r + 8*(l>>4), N = l&15).
  float* scr = &s_ac[w][0];
#pragma unroll
  for (int f = 0; f < 8; ++f) {
#pragma unroll
    for (int r = 0; r < 8; ++r)
      scr[f * 256 + l * 8 + r] = D[f][r];
  }
  const int lsrc = (h < 8) ? h : (h + 16);
  const int rr   = (h < 8) ? h : (h - 8);
  const int didx = lsrc * 8 + rr;
  // lane l writes cols {g0+2j}*16+h  -> 4 fully-coalesced 128B row stores
#pragma unroll
  for (int j = 0; j < 4; ++j) {
    const int f = g0 + 2 * j;
    const float val = scr[f * 256 + didx];
    __builtin_nontemporal_store(val, out_t + f * 16 + h);
  }
#else
  // VALU fallback: lane covers cols {l, l+32, l+64, l+96}
  float acc0 = 0.f, acc1 = 0.f, acc2 = 0.f, acc3 = 0.f;
#pragma unroll
  for (int k = 0; k < 32; ++k) {
    const float a = s_acf[w][k * 16 + h] * rn;
    acc0 += s_nd[w][k * 128 + l]      * a;
    acc1 += s_nd[w][k * 128 + 32 + l] * a;
    acc2 += s_nd[w][k * 128 + 64 + l] * a;
    acc3 += s_nd[w][k * 128 + 96 + l] * a;
  }
  __builtin_nontemporal_store(acc0, out_t + l);
  __builtin_nontemporal_store(acc1, out_t + 32 + l);
  __builtin_nontemporal_store(acc2, out_t + 64 + l);
  __builtin_nontemporal_store(acc3, out_t + 96 + l);
#endif
}

extern "C" void kernel_launch(void* const* d_in, const int* in_sizes, int n_in,
                              void* d_out, int out_size, void* d_ws, size_t ws_size,
                              hipStream_t stream) {
  (void)in_sizes; (void)n_in; (void)out_size; (void)d_ws; (void)ws_size;
  const float* beta  = (const float*)d_in[0];
  const float* selfa = (const float*)d_in[1];
  const float* ac    = (const float*)d_in[2];
  const float* nodes = (const float*)d_in[3];
  const float* gw    = (const float*)d_in[4];
  float* out = (float*)d_out;                       // [16384,128]
  float* acf = out + (size_t)8 * 2048 * 128;        // [16384,32,16]
  const int tokens = 8 * 2048;
  attn_fused<<<tokens / 2, 64, 0, stream>>>(beta, selfa, ac, nodes, gw, out, acf);
}